// GraphConvLayer_4535485464578
// MI455X (gfx1250) — compile-verified
//
#include <hip/hip_runtime.h>

typedef __attribute__((ext_vector_type(2))) float v2f;
typedef __attribute__((ext_vector_type(8))) float v8f;

#define N_NODES 543
#define FEAT    256
#define OUT_D   256
#define NPAD    544   // 34 tiles of 16
#define DQ      64    // output-column slice per workgroup
#define NWGS    4096  // 1024 (b*t) * 4 d-quarters

extern "C" __global__ __launch_bounds__(256)
void gcn_fused_wmma(const float* __restrict__ x,
                    const float* __restrict__ W,
                    const float* __restrict__ bias,
                    const float* __restrict__ A,
                    float* __restrict__ out)
{
    extern __shared__ float s_sup[];   // [NPAD][DQ] = 544*64 floats = 136 KB

    const int bt = blockIdx.x >> 2;    // which (b,t) slice
    const int dq = blockIdx.x & 3;     // which 64-col slice of OUT
    const int d0 = dq * DQ;

    const int lane = threadIdx.x & 31;
    const int wave = threadIdx.x >> 5; // 0..7
    const int l15  = lane & 15;
    const int hi   = lane >> 4;        // 0 for lanes 0-15, 1 for 16-31

    const int nt = wave & 3;           // n-tile within the 64-col slice
    const int mh = wave >> 2;          // 0..1 : which half of the 34 m-tiles
    const int n0 = nt * 16;

    const float* __restrict__ xbt = x   + (size_t)bt * N_NODES * FEAT;
    float*       __restrict__ obt = out + (size_t)bt * N_NODES * OUT_D;

    const float bv = bias[d0 + n0 + l15];

    // ---------------- Phase 1: S = X * W + b  (cols d0..d0+63) ----------------
    // Row 543 is padding: clamp the address (no branch). Its S values are
    // finite garbage, and phase 2 multiplies them by an explicit 0.
    for (int mi = 0; mi < 17; ++mi) {
        const int m0   = (mh * 17 + mi) * 16;
        const int arow = m0 + l15;
        const int xrow = (arow < N_NODES) ? arow : (N_NODES - 1);
        const float* __restrict__ xr = xbt + (size_t)xrow * FEAT + 2 * hi; // 8B aligned
        const float* __restrict__ wc = W + (size_t)(2 * hi) * OUT_D + d0 + n0 + l15;

        v8f acc = {0.f, 0.f, 0.f, 0.f, 0.f, 0.f, 0.f, 0.f};
        #pragma unroll 8
        for (int k = 0; k < FEAT; k += 4) {
            // A-frag (16x4 f32): lane holds x[row][k+2*hi], x[row][k+2*hi+1]
            const float2 av = *(const float2*)(xr + k);
            v2f a; a.x = av.x; a.y = av.y;
            // B-frag (4x16 f32): VGPR0 = W[k+2*hi][col], VGPR1 = W[k+2*hi+1][col]
            v2f b;
            b.x = wc[(size_t)k * OUT_D];
            b.y = wc[(size_t)(k + 1) * OUT_D];
            acc = __builtin_amdgcn_wmma_f32_16x16x4_f32(
                false, a, false, b, (short)0, acc, false, false);
        }
        // C/D layout: VGPR v -> row m0+v+8*hi, lane -> col n0+l15
        #pragma unroll
        for (int v = 0; v < 8; ++v) {
            const int r = m0 + v + 8 * hi;
            s_sup[r * DQ + n0 + l15] = acc[v] + bv;
        }
    }

    __syncthreads();

    // ---------------- Phase 2: O = A * S, ReLU ----------------
    for (int mi = 0; mi < 17; ++mi) {
        const int m0   = (mh * 17 + mi) * 16;
        const int arow = m0 + l15;
        const int acrow = (arow < N_NODES) ? arow : (N_NODES - 1); // clamped; row 543 never stored
        const float* __restrict__ Ar = A + (size_t)acrow * N_NODES;

        v8f acc = {0.f, 0.f, 0.f, 0.f, 0.f, 0.f, 0.f, 0.f};

        // Main loop: k = 0..536 -> c0 <= 538, c0+1 <= 539 : fully in-bounds.
        #pragma unroll 5
        for (int k = 0; k < 540; k += 4) {
            const int c0 = k + 2 * hi;
            v2f a;
            a.x = Ar[c0];
            a.y = Ar[c0 + 1];
            v2f b;
            b.x = s_sup[(c0    ) * DQ + n0 + l15];
            b.y = s_sup[(c0 + 1) * DQ + n0 + l15];
            acc = __builtin_amdgcn_wmma_f32_16x16x4_f32(
                false, a, false, b, (short)0, acc, false, false);
        }
        // Peeled last step k=540: c0 = 540+2*hi (<=542 ok); c0+1 hits 543 for hi lanes.
        {
            const int c0 = 540 + 2 * hi;
            const float ay_raw = Ar[(c0 + 1 < N_NODES) ? (c0 + 1) : (N_NODES - 1)];
            v2f a;
            a.x = Ar[c0];
            a.y = (c0 + 1 < N_NODES) ? ay_raw : 0.f;   // value select only (cndmask)
            v2f b;
            b.x = s_sup[(c0    ) * DQ + n0 + l15];
            b.y = s_sup[(c0 + 1) * DQ + n0 + l15];     // row 543 exists in padded LDS
            acc = __builtin_amdgcn_wmma_f32_16x16x4_f32(
                false, a, false, b, (short)0, acc, false, false);
        }

        #pragma unroll
        for (int v = 0; v < 8; ++v) {
            const int r = m0 + v + 8 * hi;
            if (r < N_NODES) {
                const float val = acc[v];
                obt[(size_t)r * OUT_D + d0 + n0 + l15] = val > 0.f ? val : 0.f;
            }
        }
    }
}

extern "C" void kernel_launch(void* const* d_in, const int* in_sizes, int n_in,
                              void* d_out, int out_size, void* d_ws, size_t ws_size,
                              hipStream_t stream) {
    (void)in_sizes; (void)n_in; (void)d_ws; (void)ws_size; (void)out_size;
    const float* x    = (const float*)d_in[0];   // [16,64,543,256]
    const float* W    = (const float*)d_in[1];   // [256,256]
    const float* bias = (const float*)d_in[2];   // [256]
    const float* A    = (const float*)d_in[3];   // [543,543]
    float* out = (float*)d_out;                  // [16,64,543,256]

    const size_t lds_bytes = (size_t)NPAD * DQ * sizeof(float); // 139264 B
    gcn_fused_wmma<<<NWGS, 256, lds_bytes, stream>>>(x, W, bias, A, out);
}